// GNNModel_274877907756
// MI455X (gfx1250) — compile-verified
//
#include <hip/hip_runtime.h>

typedef __attribute__((ext_vector_type(2))) float v2f;
typedef __attribute__((ext_vector_type(4))) float v4f;
typedef __attribute__((ext_vector_type(8))) float v8f;

#define D_NODE 256
#define D_EDGE 128
#define D_OUT  128

// ---------------------------------------------------------------------------
// Zero the ax accumulator (float4 stores)
// ---------------------------------------------------------------------------
__global__ __launch_bounds__(256) void zero_f32(float* __restrict__ p, long long n4) {
    long long i = (long long)blockIdx.x * blockDim.x + threadIdx.x;
    if (i < n4) {
        v4f z = {0.f, 0.f, 0.f, 0.f};
        ((v4f*)p)[i] = z;
    }
}

// ---------------------------------------------------------------------------
// ax[dst[e]] += x[src[e]]   (segment_sum over edges)
// one thread per (edge, 4-float chunk): 64 threads per edge, float4 loads,
// relaxed agent-scope float atomic adds -> global_atomic_add_f32
// ---------------------------------------------------------------------------
__global__ __launch_bounds__(256) void scatter_add_rows(
    const float* __restrict__ x,
    const long long* __restrict__ src,
    const long long* __restrict__ dst,
    float* __restrict__ ax, int E) {
    long long tid = (long long)blockIdx.x * blockDim.x + threadIdx.x;
    int e = (int)(tid >> 6);
    if (e >= E) return;
    int c = ((int)tid & 63) * 4;
    long long s = src[e];
    long long d = dst[e];
    v4f v = *(const v4f*)(x + (long long)s * D_NODE + c);
    float* o = ax + (long long)d * D_NODE + c;
    __hip_atomic_fetch_add(o + 0, v.x, __ATOMIC_RELAXED, __HIP_MEMORY_SCOPE_AGENT);
    __hip_atomic_fetch_add(o + 1, v.y, __ATOMIC_RELAXED, __HIP_MEMORY_SCOPE_AGENT);
    __hip_atomic_fetch_add(o + 2, v.z, __ATOMIC_RELAXED, __HIP_MEMORY_SCOPE_AGENT);
    __hip_atomic_fetch_add(o + 3, v.w, __ATOMIC_RELAXED, __HIP_MEMORY_SCOPE_AGENT);
}

// ---------------------------------------------------------------------------
// C[M,128] = A[M,K] @ B[K,128] with V_WMMA_F32_16X16X4_F32 (wave32).
// One wave owns one 16x16 output tile; 8 waves/block span the 128-wide N.
// M must be a multiple of 16 (6250 tiles / 40000 tiles here -> exact).
// Fragment layouts per ISA 7.12.2 (32-bit A 16x4 / B,C,D row-striped):
//   A: lane l (half = l>>4, r = l&15) holds A[m0+r][k + 2*half + {0,1}]
//   B: lane l holds B[k + 2*half + {0,1}][n0 + r] in VGPR {0,1}
//   C/D: VGPR j holds C[m0 + j + 8*half][n0 + r]
// ---------------------------------------------------------------------------
template <int K>
__global__ __launch_bounds__(256) void gemm_n128_wmma_f32(
    const float* __restrict__ A,
    const float* __restrict__ B,
    float* __restrict__ C) {
    const int lane = threadIdx.x & 31;
    const int wave = threadIdx.x >> 5;
    const int half = lane >> 4;   // 0 or 1
    const int r    = lane & 15;
    const int m0   = blockIdx.x * 16;
    const int n0   = wave * 16;

    const float* arow = A + (long long)(m0 + r) * K + 2 * half;         // step +k
    const float* bcol = B + (long long)(2 * half) * D_OUT + n0 + r;     // step +k*128

    v8f acc = {};
#pragma unroll 8
    for (int k = 0; k < K; k += 4) {
        v2f a = *(const v2f*)(arow + k);                 // A[m][k+2h], A[m][k+2h+1]
        v2f b;
        b.x = bcol[(long long)k * D_OUT];                // B[k+2h  ][n]
        b.y = bcol[(long long)(k + 1) * D_OUT];          // B[k+2h+1][n]
        acc = __builtin_amdgcn_wmma_f32_16x16x4_f32(
            /*neg_a=*/false, a, /*neg_b=*/false, b,
            /*c_mod=*/(short)0, acc, /*reuse_a=*/false, /*reuse_b=*/false);
    }

    float* crow = C + (long long)(m0 + 8 * half) * D_OUT + n0 + r;
#pragma unroll
    for (int j = 0; j < 8; ++j)
        crow[(long long)j * D_OUT] = acc[j];
}

// ---------------------------------------------------------------------------
// scores[e] = (sum_d axw[src][d] * ew[e][d] * axw[dst][d]) * tc[e] * lw
// 32 lanes per edge, float4 per lane, wave32 xor-shuffle reduction.
// ---------------------------------------------------------------------------
__global__ __launch_bounds__(256) void distmult_scores(
    const float* __restrict__ axw,
    const float* __restrict__ ew,
    const long long* __restrict__ src,
    const long long* __restrict__ dst,
    const float* __restrict__ tc,
    const float* __restrict__ lw,
    float* __restrict__ scores, int E) {
    int lane = threadIdx.x & 31;
    int e = blockIdx.x * 8 + (threadIdx.x >> 5);
    if (e >= E) return;
    long long s = src[e];
    long long d = dst[e];
    int c = lane * 4;
    v4f h = *(const v4f*)(axw + (long long)s * D_OUT + c);
    v4f t = *(const v4f*)(axw + (long long)d * D_OUT + c);
    v4f w = *(const v4f*)(ew + (long long)e * D_OUT + c);
    float sum = h.x * w.x * t.x + h.y * w.y * t.y + h.z * w.z * t.z + h.w * w.w * t.w;
#pragma unroll
    for (int o = 16; o > 0; o >>= 1)
        sum += __shfl_xor(sum, o, 32);
    if (lane == 0)
        scores[e] = sum * (tc[e] * lw[0]);
}

// ---------------------------------------------------------------------------
extern "C" void kernel_launch(void* const* d_in, const int* in_sizes, int n_in,
                              void* d_out, int out_size, void* d_ws, size_t ws_size,
                              hipStream_t stream) {
    const float*     x         = (const float*)d_in[0];      // [N, 256]
    const float*     edge_attr = (const float*)d_in[1];      // [E, 128]
    const float*     tc        = (const float*)d_in[2];      // [E]
    const float*     w_node    = (const float*)d_in[3];      // [256, 128]
    const float*     w_edge    = (const float*)d_in[4];      // [128, 128]
    const float*     lw        = (const float*)d_in[5];      // [1]
    const long long* ei        = (const long long*)d_in[6];  // [2, E]

    const int N = in_sizes[0] / D_NODE;  // 100000
    const int E = in_sizes[2];           // 640000

    float* ax  = (float*)d_ws;                    // [N, 256] scratch accumulator
    float* axw = (float*)d_out;                   // [N, 128]
    float* ew  = axw + (long long)N * D_OUT;      // [E, 128]
    float* sc  = ew + (long long)E * D_OUT;       // [E]

    const long long* src = ei;
    const long long* dst = ei + E;

    // 1) ax = 0
    long long n4 = (long long)N * D_NODE / 4;
    zero_f32<<<(unsigned)((n4 + 255) / 256), 256, 0, stream>>>(ax, n4);

    // 2) ax[dst] += x[src]
    long long st = (long long)E * 64;
    scatter_add_rows<<<(unsigned)((st + 255) / 256), 256, 0, stream>>>(x, src, dst, ax, E);

    // 3) axw = ax @ w_node    (M = N, K = 256)
    gemm_n128_wmma_f32<D_NODE><<<N / 16, 256, 0, stream>>>(ax, w_node, axw);

    // 4) ew = edge_attr @ w_edge    (M = E, K = 128)
    gemm_n128_wmma_f32<D_EDGE><<<E / 16, 256, 0, stream>>>(edge_attr, w_edge, ew);

    // 5) DistMult scores
    distmult_scores<<<(E + 7) / 8, 256, 0, stream>>>(axw, ew, src, dst, tc, lw, sc, E);
}